// CTCLossWrapper_22711787061788
// MI455X (gfx1250) — compile-verified
//
#include <hip/hip_runtime.h>

// Problem constants from the reference: B=64, T=1000, D=512, V=29, L=200
#define B_ 64
#define T_ 1000
#define D_ 512
#define V_ 29
#define L_ 200
#define S_ 401        // 2*L+1
#define BT_ 64000     // B*T rows of the GEMM
#define VP_ 32        // V padded to 32 for WMMA N-tiling
#define KSTR 130      // padded per-column K stride in LDS (even -> b64 align,
                      // 130%64 spreads the 16 reading lanes across banks)
#define NEGC (-1e9f)
#define BLANK_ 28

typedef __attribute__((ext_vector_type(2))) float v2f;
typedef __attribute__((ext_vector_type(8))) float v8f;

__global__ void init_out_kernel(float* out) { out[0] = 0.0f; }

// logits[row, 0:32] = features[row, :] @ W[:, 0:29 (pad 0)] + bias
// Block: 128 threads = 4 waves; each wave -> 16 rows x 32 cols via
// V_WMMA_F32_16X16X4_F32. Grid: 64000/64 = 1000 blocks.
// W is staged TRANSPOSED in LDS so each B fragment is one ds_load_b64.
__global__ void __launch_bounds__(128)
gemm_logits_kernel(const float* __restrict__ F, const float* __restrict__ W,
                   const float* __restrict__ bias, float* __restrict__ logits) {
  __shared__ float sWt[VP_ * KSTR];  // [col][k] 16.25 KB K-chunk of padded W^T
  const int tid  = threadIdx.x;
  const int wave = tid >> 5;
  const int lane = tid & 31;
  const int half = lane >> 4;   // ISA A-layout: half-wave selects K pair
  const int l15  = lane & 15;   // ISA A-layout: M = lane&15 ; B/D: N = lane&15
  const int m0   = blockIdx.x * 64 + wave * 16;
  const float* arow = F + (size_t)(m0 + l15) * D_;

  v8f acc0 = {0.f,0.f,0.f,0.f,0.f,0.f,0.f,0.f};
  v8f acc1 = {0.f,0.f,0.f,0.f,0.f,0.f,0.f,0.f};

  const int c0 = l15, c1 = 16 + l15;

  for (int kb = 0; kb < D_; kb += 128) {
    // stage W chunk -> LDS transposed: sWt[c*KSTR + r] = W[kb+r][c] (0-pad)
    for (int i = tid; i < 128 * VP_; i += 128) {
      int r = i >> 5, c = i & 31;
      sWt[c * KSTR + r] = (c < V_) ? W[(size_t)(kb + r) * V_ + c] : 0.0f;
    }
    __syncthreads();
#pragma unroll
    for (int kk = 0; kk < 128; kk += 4) {
      const int ko = kk + 2 * half;
      v2f a;                         // A 16x4 f32: v0=K0/K2, v1=K1/K3
      a.x = arow[kb + ko];
      a.y = arow[kb + ko + 1];
      // B 4x16 f32 fragment: consecutive k -> contiguous in sWt: one b64 load
      const v2f b0 = *(const v2f*)&sWt[c0 * KSTR + ko];
      const v2f b1 = *(const v2f*)&sWt[c1 * KSTR + ko];
      acc0 = __builtin_amdgcn_wmma_f32_16x16x4_f32(
          false, a, false, b0, (short)0, acc0, false, false);
      acc1 = __builtin_amdgcn_wmma_f32_16x16x4_f32(
          false, a, false, b1, (short)0, acc1, false, false);
    }
    __syncthreads();
  }

  const float bv0 = bias[c0];                      // c0 <= 15 < 29
  const float bv1 = (c1 < V_) ? bias[c1] : 0.0f;
#pragma unroll
  for (int r = 0; r < 8; ++r) {                    // D layout: M = r + 8*half
    const int row = m0 + r + 8 * half;
    logits[(size_t)row * VP_ + c0] = acc0[r] + bv0;
    logits[(size_t)row * VP_ + c1] = acc1[r] + bv1;
  }
}

// In-place log_softmax over the valid 29 columns of each padded 32-wide row.
__global__ void __launch_bounds__(256)
logsoftmax_kernel(float* __restrict__ lp) {
  const int row = blockIdx.x * blockDim.x + threadIdx.x;
  if (row >= BT_) return;
  float* p = lp + (size_t)row * VP_;
  float x[V_];
  float m = NEGC;
#pragma unroll
  for (int j = 0; j < V_; ++j) { x[j] = p[j]; m = fmaxf(m, x[j]); }
  float s = 0.0f;
#pragma unroll
  for (int j = 0; j < V_; ++j) s += expf(x[j] - m);
  const float lse = m + logf(s);
#pragma unroll
  for (int j = 0; j < V_; ++j) p[j] = x[j] - lse;
  p[29] = NEGC; p[30] = NEGC; p[31] = NEGC;
}

// CTC forward trellis: one block per batch element, thread = extended state s.
// alpha double-buffered in LDS; one barrier per timestep; 1000 steps.
__global__ void __launch_bounds__(416)
ctc_alpha_kernel(const float* __restrict__ lp, const int* __restrict__ labels,
                 const int* __restrict__ flen, const int* __restrict__ llen,
                 float* __restrict__ out) {
  const int b = blockIdx.x;
  const int s = threadIdx.x;                 // 0..415, states 0..400 valid
  __shared__ float alpha[2][416];

  int  ext  = BLANK_;
  bool skip = false;
  if (s < S_ && (s & 1)) {
    ext  = labels[b * L_ + (s >> 1)];
    skip = (s >= 2) && (ext != labels[b * L_ + (s >> 1) - 1]);
  }
  const int fl = flen[b];
  const float* lpb = lp + (size_t)b * T_ * VP_;

  float a0 = NEGC;
  if (s == 0)      a0 = lpb[BLANK_];
  else if (s == 1) a0 = lpb[ext];
  alpha[0][s] = a0;
  __syncthreads();

  int cur = 0;
  for (int t = 1; t < T_; ++t) {
    const float a1 = alpha[cur][s];
    float nv = a1;
    if (s < S_) {
      const float a2 = (s >= 1) ? alpha[cur][s - 1] : NEGC;
      const float a3 = skip     ? alpha[cur][s - 2] : NEGC;
      const float m  = fmaxf(a1, fmaxf(a2, a3));
      const float sm = expf(a1 - m) + expf(a2 - m) + expf(a3 - m);
      const float lpt = lpb[t * VP_ + ext];
      if (t < fl) nv = lpt + m + logf(sm);   // freeze when t >= feature_len
    }
    alpha[cur ^ 1][s] = nv;
    __syncthreads();
    cur ^= 1;
  }

  if (s == 0) {
    const int ll  = llen[b];
    const int idx = 2 * ll;
    const float l1 = alpha[cur][idx];
    const float l2 = (ll > 0) ? alpha[cur][idx - 1] : NEGC;
    const float m  = fmaxf(l1, l2);
    float nll = -(m + logf(expf(l1 - m) + expf(l2 - m)));
    const float denom = (float)(ll > 0 ? ll : 1);
    nll = (nll < 5e8f) ? (nll / denom) : 0.0f;   // zero_infinity + /len
    atomicAdd(out, nll * (1.0f / (float)B_));    // mean over batch
  }
}

extern "C" void kernel_launch(void* const* d_in, const int* in_sizes, int n_in,
                              void* d_out, int out_size, void* d_ws, size_t ws_size,
                              hipStream_t stream) {
  const float* F     = (const float*)d_in[0];  // [B,T,D]
  const float* W     = (const float*)d_in[1];  // [D,V]
  const float* bias  = (const float*)d_in[2];  // [V]
  const int* labels  = (const int*)d_in[3];    // [B,L]
  const int* flen    = (const int*)d_in[4];    // [B]
  const int* llen    = (const int*)d_in[5];    // [B]
  float* out    = (float*)d_out;
  float* logits = (float*)d_ws;                // [B*T, 32] = 8.2 MB scratch

  init_out_kernel<<<1, 1, 0, stream>>>(out);
  gemm_logits_kernel<<<BT_ / 64, 128, 0, stream>>>(F, W, bias, logits);
  logsoftmax_kernel<<<(BT_ + 255) / 256, 256, 0, stream>>>(logits);
  ctc_alpha_kernel<<<B_, 416, 0, stream>>>(logits, labels, flen, llen, out);
}